// LayerList_43284680409770
// MI455X (gfx1250) — compile-verified
//
#include <hip/hip_runtime.h>
#include <stdint.h>

typedef __attribute__((ext_vector_type(2))) float v2f;
typedef __attribute__((ext_vector_type(4))) float v4f;
typedef __attribute__((ext_vector_type(8))) float v8f;

#define BATCH   8192
#define DIM     4096
#define ROWLEN  (DIM * 2)          // 8192 floats per batch row
#define QPR     (ROWLEN / 4)       // 2048 float4 quads per row
#define TPB     256                // threads per block = quads per block
#define UPB     (TPB * 2)          // 512 units covered per block
#define GRIDX   (QPR / TPB)        // 8 column blocks
#define GRIDY   512
#define ROWS_PB (BATCH / GRIDY)    // 16 rows per block

// ---------------------------------------------------------------------------
// Kernel 1: batched 2x2 matmul W = W1[u] @ W2[u] on the WMMA pipe.
// 8 units packed block-diagonally into one V_WMMA_F32_16X16X4_F32:
//   unit i: A rows 2i..2i+1, K pair (i%2)*2 ; B K pair (i%2)*2, N cols 2i..2i+1
// Cross-terms land off the diagonal blocks and are ignored. Output per unit:
//   a = W00-W01, b = W10-W11  (sign-test coefficients; softmax+argmax folds
//   to sign(x0*a + x1*b)).
// ---------------------------------------------------------------------------
#define FOLD_TPB        256
#define WAVES_PER_BLK   (FOLD_TPB / 32)                 // 8 waves
#define UNITS_PER_WAVE  8
#define UNITS_PER_FBLK  (WAVES_PER_BLK * UNITS_PER_WAVE) // 64 units/block

__global__ __launch_bounds__(FOLD_TPB) void fold_coef_wmma_kernel(
    const float* __restrict__ W1, const float* __restrict__ W2,
    float* __restrict__ coef) {
  __shared__ float lds_d[WAVES_PER_BLK][16][16];        // 8 KB

  const int tid  = threadIdx.x;
  const int wave = tid >> 5;
  const int lane = tid & 31;
  const int half = lane >> 4;       // 0: lanes 0-15, 1: lanes 16-31
  const int lm   = lane & 15;       // M (A / D) or N (B / D)
  const int ub   = (blockIdx.x * WAVES_PER_BLK + wave) * UNITS_PER_WAVE;

  // A operand (16x4 f32): lane -> M=lm, VGPR v -> K = 2*half + v.
  // A[m][k] = W1[i][r][v] iff half == (i&1), with i=m/2, r=m%2.
  const int ia = lm >> 1, ra = lm & 1;
  float a0 = 0.0f, a1 = 0.0f;
  if (half == (ia & 1)) {
    const float* w1 = W1 + (ub + ia) * 4 + ra * 2;  // W1[u] row-major (in,out)
    a0 = w1[0]; a1 = w1[1];
  }
  // B operand (4x16 f32): lane -> N=lm, VGPR v -> K = v + 2*half.
  // B[k][n] = W2[j][v][c] iff half == (j&1), with j=n/2, c=n%2.
  const int jb = lm >> 1, cb = lm & 1;
  float b0 = 0.0f, b1 = 0.0f;
  if (half == (jb & 1)) {
    const float* w2 = W2 + (ub + jb) * 4 + cb;      // W2[j][v][c] = w2[v*2]
    b0 = w2[0]; b1 = w2[2];
  }

  v2f A = {a0, a1};
  v2f B = {b0, b1};
  v8f C = {0.f, 0.f, 0.f, 0.f, 0.f, 0.f, 0.f, 0.f};
  // (neg_a, A, neg_b, B, c_mod, C, reuse_a, reuse_b)
  v8f D = __builtin_amdgcn_wmma_f32_16x16x4_f32(false, A, false, B,
                                                (short)0, C, false, false);

  // D layout: VGPR g, lane -> M = g + 8*half, N = lm. Spill tile to LDS.
#pragma unroll
  for (int g = 0; g < 8; ++g)
    lds_d[wave][g + 8 * half][lm] = D[g];
  __syncthreads();

  // Lanes 0..15 of each wave emit the two coefficients of its 8 units:
  // coef[u*2+r] = D[2i+r][2i] - D[2i+r][2i+1]
  if (lane < 16) {
    const int i = lane >> 1, r = lane & 1;
    const int m = 2 * i + r;
    float val = lds_d[wave][m][2 * i] - lds_d[wave][m][2 * i + 1];
    coef[(ub + i) * 2 + r] = val;
  }
}

// ---------------------------------------------------------------------------
// Kernel 2: stream 512 MB at HBM rate. Coefs staged to LDS via the CDNA5
// async-to-LDS path; x/out use non-temporal 128-bit accesses.
// ---------------------------------------------------------------------------
__global__ __launch_bounds__(TPB) void stream_decide_kernel(
    const float* __restrict__ x, const float* __restrict__ coef,
    float* __restrict__ out) {
  __shared__ float lds_coef[UPB * 2];              // 512 units * (a,b) = 4 KB

  const int t        = threadIdx.x;
  const int colQuad  = blockIdx.x * TPB + t;       // float4 column in the row
  const int unitBase = blockIdx.x * UPB;           // first unit of this block

  // --- CDNA5 async copy: this block's coefficient slice -> LDS (16 B/lane) ---
  {
    const float* gsrc = coef + unitBase * 2 + t * 4;
    unsigned lds_b = (unsigned)(uintptr_t)lds_coef + (unsigned)(t * 16);
    asm volatile("global_load_async_to_lds_b128 %0, %1, off"
                 :: "v"(lds_b), "v"(gsrc) : "memory");
#if __has_builtin(__builtin_amdgcn_s_wait_asynccnt)
    __builtin_amdgcn_s_wait_asynccnt(0);
#else
    asm volatile("s_wait_asynccnt 0" ::: "memory");
#endif
  }
  __syncthreads();

  // Thread t's quad covers units 2t, 2t+1 of this block: LDS float4 = (a0,b0,a1,b1)
  const v4f c  = ((const v4f*)lds_coef)[t];
  const float a0 = c.x, b0 = c.y, a1 = c.z, b1 = c.w;

  const long base = (long)blockIdx.y * ROWS_PB * (long)ROWLEN + (long)colQuad * 4;
  const float* __restrict__ xp = x + base;
  float* __restrict__ op = out + base;

#pragma unroll 2
  for (int i = 0; i < ROWS_PB; ++i) {
    v4f xv = __builtin_nontemporal_load((const v4f*)xp);
    if (i + 1 < ROWS_PB) __builtin_prefetch(xp + ROWLEN, 0, 0);
    float s0 = xv.x * a0 + xv.y * b0;
    float s1 = xv.z * a1 + xv.w * b1;
    v4f o;
    o.x = (s0 >= 0.0f) ? 1.0f : 0.0f;   // argmax ties -> index 0, matches jnp.argmax
    o.y = 1.0f - o.x;
    o.z = (s1 >= 0.0f) ? 1.0f : 0.0f;
    o.w = 1.0f - o.z;
    __builtin_nontemporal_store(o, (v4f*)op);
    xp += ROWLEN;
    op += ROWLEN;
  }
}

// ---------------------------------------------------------------------------
extern "C" void kernel_launch(void* const* d_in, const int* in_sizes, int n_in,
                              void* d_out, int out_size, void* d_ws, size_t ws_size,
                              hipStream_t stream) {
  const float* x  = (const float*)d_in[0];
  const float* W1 = (const float*)d_in[1];
  const float* W2 = (const float*)d_in[2];
  float* out  = (float*)d_out;
  float* coef = (float*)d_ws;   // needs DIM*2*4 = 32 KB of scratch

  fold_coef_wmma_kernel<<<DIM / UNITS_PER_FBLK, FOLD_TPB, 0, stream>>>(W1, W2, coef);

  dim3 grid(GRIDX, GRIDY);
  stream_decide_kernel<<<grid, TPB, 0, stream>>>(x, coef, out);
}